// SRenderY_74191265071384
// MI455X (gfx1250) — compile-verified
//
#include <hip/hip_runtime.h>
#include <hip/hip_bf16.h>

#define Bc 4
#define Vn 5023
#define Fn 9976
#define Hh 224
#define Ww 224
#define UVn 256
#define NC 624            // ceil(F/16) face chunks
#define FP (NC * 16)      // padded face count
#define CSTRIDE 192       // floats per (b,chunk) coef block: 3 q * 64 -> 768B
#define TILES_PER_B ((Hh * Ww) / 16)      // 3136 tiles of 16 pixels (16x1)
#define BLOCKS_PER_B (TILES_PER_B / 8)    // 392 blocks, 8 waves each

typedef __attribute__((ext_vector_type(2))) float v2f;
typedef __attribute__((ext_vector_type(8))) float v8f;

// ---------------------------------------------------------------- utilities
__global__ void zero_f(float* p, int n) {
    int i = blockIdx.x * blockDim.x + threadIdx.x;
    if (i < n) p[i] = 0.0f;
}

// ------------------------------------------------- vertex normal scatter-add
__global__ void nrm_scatter(const float* __restrict__ verts,
                            const int* __restrict__ faces,
                            float* __restrict__ nacc) {
    int i = blockIdx.x * blockDim.x + threadIdx.x;
    if (i >= Bc * Fn) return;
    int b = i / Fn, f = i % Fn;
    int i0 = faces[f * 3 + 0], i1 = faces[f * 3 + 1], i2 = faces[f * 3 + 2];
    const float* vb = verts + (size_t)b * Vn * 3;
    float ax = vb[i1*3+0] - vb[i0*3+0];
    float ay = vb[i1*3+1] - vb[i0*3+1];
    float az = vb[i1*3+2] - vb[i0*3+2];
    float bx = vb[i2*3+0] - vb[i0*3+0];
    float by = vb[i2*3+1] - vb[i0*3+1];
    float bz = vb[i2*3+2] - vb[i0*3+2];
    float cx = ay * bz - az * by;
    float cy = az * bx - ax * bz;
    float cz = ax * by - ay * bx;
    float* nb = nacc + (size_t)b * Vn * 3;
    atomicAdd(&nb[i0*3+0], cx); atomicAdd(&nb[i0*3+1], cy); atomicAdd(&nb[i0*3+2], cz);
    atomicAdd(&nb[i1*3+0], cx); atomicAdd(&nb[i1*3+1], cy); atomicAdd(&nb[i1*3+2], cz);
    atomicAdd(&nb[i2*3+0], cx); atomicAdd(&nb[i2*3+1], cy); atomicAdd(&nb[i2*3+2], cz);
}

__global__ void nrm_norm(float* __restrict__ nacc) {
    int i = blockIdx.x * blockDim.x + threadIdx.x;
    if (i >= Bc * Vn) return;
    float x = nacc[i*3+0], y = nacc[i*3+1], z = nacc[i*3+2];
    float n = fmaxf(sqrtf(x*x + y*y + z*z), 1e-6f);
    nacc[i*3+0] = x / n; nacc[i*3+1] = y / n; nacc[i*3+2] = z / n;
}

// --------------------------------------------- per-face affine coefficients
// Per (b, chunk): 192 floats = 3 quantities x 64. Within a quantity block,
// entry pair 2L holds the WMMA B-operand b64 for lane L:
//   L in [0,16):  (a_L, b_L)        (K=0,1 of column L)
//   L in [16,32): (c_{L-16}, 0.0)   (K=2,3 of column L-16)
__global__ void coef_kernel(const float* __restrict__ tverts,
                            const int* __restrict__ faces,
                            float* __restrict__ coef) {
    int i = blockIdx.x * blockDim.x + threadIdx.x;
    if (i >= Bc * FP) return;
    int b = i / FP, f = i % FP;
    int c = f >> 4, fi = f & 15;
    float A0x = 0.f, A0y = 0.f, A0c = -1.f;   // never-inside default
    float A1x = 0.f, A1y = 0.f, A1c = 0.f;
    float Zx = 0.f,  Zy = 0.f,  Zc = 0.f;
    if (f < Fn) {
        int i0 = faces[f*3+0], i1 = faces[f*3+1], i2 = faces[f*3+2];
        const float* tb = tverts + (size_t)b * Vn * 3;
        float x0 = (tb[i0*3+0] + 1.f) * (Ww * 0.5f) - 0.5f;
        float y0 = (tb[i0*3+1] + 1.f) * (Hh * 0.5f) - 0.5f;
        float z0 = (tb[i0*3+2] + 10.f) * (Ww * 0.5f);
        float x1 = (tb[i1*3+0] + 1.f) * (Ww * 0.5f) - 0.5f;
        float y1 = (tb[i1*3+1] + 1.f) * (Hh * 0.5f) - 0.5f;
        float z1 = (tb[i1*3+2] + 10.f) * (Ww * 0.5f);
        float x2 = (tb[i2*3+0] + 1.f) * (Ww * 0.5f) - 0.5f;
        float y2 = (tb[i2*3+1] + 1.f) * (Hh * 0.5f) - 0.5f;
        float z2 = (tb[i2*3+2] + 10.f) * (Ww * 0.5f);
        float d = (y1 - y2) * (x0 - x2) + (x2 - x1) * (y0 - y2);
        if (fabsf(d) > 1e-8f) {
            float inv = 1.0f / d;
            A0x = (y1 - y2) * inv;
            A0y = (x2 - x1) * inv;
            A0c = -((y1 - y2) * x2 + (x2 - x1) * y2) * inv;
            A1x = (y2 - y0) * inv;
            A1y = (x0 - x2) * inv;
            A1c = -((y2 - y0) * x2 + (x0 - x2) * y2) * inv;
            float iz0 = 1.0f / z0, iz1 = 1.0f / z1, iz2 = 1.0f / z2;
            float u = iz0 - iz2, v = iz1 - iz2;
            Zx = A0x * u + A1x * v;
            Zy = A0y * u + A1y * v;
            Zc = A0c * u + A1c * v + iz2;
        }
    }
    float* base = coef + (size_t)(b * NC + c) * CSTRIDE;
    float qa[3] = {A0x, A1x, Zx};
    float qb[3] = {A0y, A1y, Zy};
    float qc[3] = {A0c, A1c, Zc};
#pragma unroll
    for (int q = 0; q < 3; ++q) {
        float* qp = base + q * 64;
        qp[fi*2 + 0]      = qa[q];
        qp[fi*2 + 1]      = qb[q];
        qp[32 + fi*2 + 0] = qc[q];
        qp[32 + fi*2 + 1] = 0.0f;
    }
}

// ---------------------------------------- async chunk staging (wave 0 only)
// Copies one 768B coef chunk global->LDS with 6 full-EXEC async b32 ops.
__device__ __forceinline__ void issue_chunk(const float* gbase, int chunk,
                                            const float* ldsbuf, int lane) {
    unsigned goff  = (unsigned)(chunk * (CSTRIDE * 4) + lane * 4);
    unsigned laddr = (unsigned)(uintptr_t)ldsbuf + (unsigned)(lane * 4);
#pragma unroll
    for (int k = 0; k < 6; ++k) {
        asm volatile("global_load_async_to_lds_b32 %0, %1, %2"
                     :: "v"(laddr + (unsigned)(k * 128)),
                        "v"(goff + (unsigned)(k * 128)),
                        "s"(gbase)
                     : "memory");
    }
}

// --------------------------------------------------------- WMMA rasterizer
// One wave32 per 16x1 pixel tile; 8 waves/block share one batch and the same
// chunk sequence, staged through double-buffered LDS by async copies.
// Winner kept as (hi=zl bits, lo=fid+1): hi>bhi strict update keeps earliest
// face on exact-depth ties; lo==0 means no hit -> tri = lo-1 = -1.
__global__ __launch_bounds__(256) void raster_kernel(const float* __restrict__ coef,
                                                     int* __restrict__ tri) {
    __shared__ __align__(16) float buf[2][CSTRIDE];
    int wave = threadIdx.x >> 5;
    int lane = threadIdx.x & 31;
    int b   = blockIdx.x / BLOCKS_PER_B;                    // scalar (SGPR)
    int tb  = (blockIdx.x % BLOCKS_PER_B) * 8 + wave;
    int row = tb / (Ww / 16);
    int x0  = (tb % (Ww / 16)) * 16;
    int  m  = lane & 15;
    bool lo = lane < 16;

    const float* cb = coef + (size_t)b * NC * CSTRIDE;      // uniform base

    if (wave == 0) {
        issue_chunk(cb, 0, &buf[0][0], lane);
        issue_chunk(cb, 1, &buf[1][0], lane);
    }

    // A-operand rows = pixels [x, y, 1, 0]; lanes 0-15 carry K=0,1 ; 16-31 K=2,3
    v2f A;
    A.x = lo ? (float)(x0 + m) : 1.0f;
    A.y = lo ? (float)row      : 0.0f;

    unsigned bhi[8], blo[8];
#pragma unroll
    for (int r = 0; r < 8; ++r) { bhi[r] = 0u; blo[r] = 0u; }

    for (int c = 0; c < NC; ++c) {
        if (wave == 0) asm volatile("s_wait_asynccnt 0x6" ::: "memory");
        __syncthreads();                                    // chunk c resident

        const float* cc = &buf[c & 1][0];
        v2f B0 = *(const v2f*)(cc + 0 * 64 + lane * 2);
        v2f B1 = *(const v2f*)(cc + 1 * 64 + lane * 2);
        v2f B2 = *(const v2f*)(cc + 2 * 64 + lane * 2);

        v8f z8 = {0.f, 0.f, 0.f, 0.f, 0.f, 0.f, 0.f, 0.f};
        v8f C0 = __builtin_amdgcn_wmma_f32_16x16x4_f32(false, A, false, B0, (short)0, z8, false, false);
        v8f C1 = __builtin_amdgcn_wmma_f32_16x16x4_f32(false, A, false, B1, (short)0, z8, false, false);
        v8f C2 = __builtin_amdgcn_wmma_f32_16x16x4_f32(false, A, false, B2, (short)0, z8, false, false);

        unsigned keylo = (unsigned)(c * 16 + m + 1);
#pragma unroll
        for (int r = 0; r < 8; ++r) {
            float e0 = C0[r], e1 = C1[r];
            float e2 = 1.0f - e0 - e1;
            float mn = fminf(fminf(e0, e1), e2);
            bool inside = mn >= 0.0f;
            unsigned hi   = inside ? __float_as_uint(C2[r]) : 0u;
            unsigned lo32 = inside ? keylo : 0u;
            bool gt = hi > bhi[r];
            bhi[r] = gt ? hi   : bhi[r];
            blo[r] = gt ? lo32 : blo[r];
        }

        __syncthreads();                                    // all done with buf[c&1]
        if (wave == 0) {
            int nc2 = (c + 2 < NC) ? (c + 2) : (NC - 1);
            issue_chunk(cb, nc2, &buf[c & 1][0], lane);
        }
    }
    if (wave == 0) asm volatile("s_wait_asynccnt 0x0" ::: "memory");

    // Cross-lane max-reduce within each 16-lane half (pixels M=r / M=r+8);
    // lexicographic: larger zl wins, equal zl -> smaller face id (smaller lo).
#pragma unroll
    for (int r = 0; r < 8; ++r) {
        unsigned hi = bhi[r], lo2 = blo[r];
#pragma unroll
        for (int mk = 1; mk < 16; mk <<= 1) {
            unsigned ohi = __shfl_xor(hi, mk, 32);
            unsigned olo = __shfl_xor(lo2, mk, 32);
            bool take = (ohi > hi) | ((ohi == hi) & (olo < lo2));
            hi  = take ? ohi : hi;
            lo2 = take ? olo : lo2;
        }
        if (lane == 0 || lane == 16) {
            int px = x0 + r + (lo ? 0 : 8);
            tri[((size_t)b * Hh + row) * Ww + px] = (int)lo2 - 1;
        }
    }
}

// ------------------------------------------------------------ shade kernel
__device__ __forceinline__ float tap(const float* ab, int b, int ch, int ix, int iy) {
    bool val = (ix >= 0) & (ix < UVn) & (iy >= 0) & (iy < UVn);
    int ixc = ix < 0 ? 0 : (ix > UVn - 1 ? UVn - 1 : ix);
    int iyc = iy < 0 ? 0 : (iy > UVn - 1 ? UVn - 1 : iy);
    float v = ab[(((size_t)b * 3 + ch) * UVn + iyc) * UVn + ixc];
    return val ? v : 0.0f;
}

__global__ void shade_kernel(const float* __restrict__ albedos,
                             const float* __restrict__ lights,
                             const int* __restrict__ faces,
                             const float* __restrict__ uvcoords,
                             const float* __restrict__ cf,
                             const float* __restrict__ nrm,
                             const float* __restrict__ coef,
                             const int* __restrict__ tri,
                             float* __restrict__ out) {
    int i = blockIdx.x * blockDim.x + threadIdx.x;
    if (i >= Bc * Hh * Ww) return;
    int b  = i / (Hh * Ww);
    int hw = i % (Hh * Ww);
    size_t oc = (size_t)b * 3 * Hh * Ww + hw;
    int id = tri[i];
    if (id < 0) {
        out[oc] = 0.f; out[oc + Hh*Ww] = 0.f; out[oc + 2*Hh*Ww] = 0.f;
        return;
    }
    // Recompute barycentrics of the winning face from its affine coefficients
    int ch = id >> 4, fi = id & 15;
    const float* base = coef + (size_t)(b * NC + ch) * CSTRIDE;
    float fx = (float)(hw % Ww), fy = (float)(hw / Ww);
    float w0 = base[fi*2 + 0]          * fx + base[fi*2 + 1]          * fy + base[32 + fi*2];
    float w1 = base[64 + fi*2 + 0]     * fx + base[64 + fi*2 + 1]     * fy + base[96 + fi*2];
    float w2 = 1.0f - w0 - w1;

    int i0 = faces[id*3+0], i1 = faces[id*3+1], i2 = faces[id*3+2];
    const float* uvf = uvcoords + (size_t)id * 9;
    float u = w0 * uvf[0] + w1 * uvf[3] + w2 * uvf[6];
    float v = w0 * uvf[1] + w1 * uvf[4] + w2 * uvf[7];
    const float* nb = nrm + (size_t)b * Vn * 3;
    float Nx = w0 * nb[i0*3+0] + w1 * nb[i1*3+0] + w2 * nb[i2*3+0];
    float Ny = w0 * nb[i0*3+1] + w1 * nb[i1*3+1] + w2 * nb[i2*3+1];
    float Nz = w0 * nb[i0*3+2] + w1 * nb[i1*3+2] + w2 * nb[i2*3+2];
    float sh[9] = {1.f, Nx, Ny, Nz, Nx*Ny, Nx*Nz, Ny*Nz, Nx*Nx - Ny*Ny, 3.f*Nz*Nz - 1.f};
    float s0 = 0.f, s1 = 0.f, s2 = 0.f;
    const float* lb = lights + b * 27;
#pragma unroll
    for (int j = 0; j < 9; ++j) {
        float sj = sh[j] * cf[j];
        s0 += lb[j*3+0] * sj;
        s1 += lb[j*3+1] * sj;
        s2 += lb[j*3+2] * sj;
    }
    float gx = (u + 1.0f) * (UVn * 0.5f) - 0.5f;
    float gy = (v + 1.0f) * (UVn * 0.5f) - 0.5f;
    float xf = floorf(gx), yf = floorf(gy);
    float wx = gx - xf, wy = gy - yf;
    int ix0 = (int)xf, iy0 = (int)yf;
    float shd[3] = {s0, s1, s2};
#pragma unroll
    for (int chn = 0; chn < 3; ++chn) {
        float a = tap(albedos, b, chn, ix0,     iy0    ) * (1.f - wx) * (1.f - wy)
                + tap(albedos, b, chn, ix0 + 1, iy0    ) * wx         * (1.f - wy)
                + tap(albedos, b, chn, ix0,     iy0 + 1) * (1.f - wx) * wy
                + tap(albedos, b, chn, ix0 + 1, iy0 + 1) * wx         * wy;
        out[oc + (size_t)chn * Hh * Ww] = a * shd[chn];
    }
}

// --------------------------------------------------------------- dispatcher
extern "C" void kernel_launch(void* const* d_in, const int* in_sizes, int n_in,
                              void* d_out, int out_size, void* d_ws, size_t ws_size,
                              hipStream_t stream) {
    const float* vertices = (const float*)d_in[0];
    const float* tverts   = (const float*)d_in[1];
    const float* albedos  = (const float*)d_in[2];
    const float* lights   = (const float*)d_in[3];
    const int*   faces    = (const int*)d_in[4];
    const float* uvcoords = (const float*)d_in[5];
    const float* cf       = (const float*)d_in[6];
    float* out = (float*)d_out;

    char* w = (char*)d_ws;
    size_t o = 0;
    float* nrm = (float*)(w + o);
    o += (((size_t)Bc * Vn * 3 * 4) + 255) / 256 * 256;
    float* coef = (float*)(w + o);
    o += (((size_t)Bc * NC * CSTRIDE * 4) + 255) / 256 * 256;
    int* tri = (int*)(w + o);

    zero_f<<<(Bc * Vn * 3 + 255) / 256, 256, 0, stream>>>(nrm, Bc * Vn * 3);
    nrm_scatter<<<(Bc * Fn + 255) / 256, 256, 0, stream>>>(vertices, faces, nrm);
    nrm_norm<<<(Bc * Vn + 255) / 256, 256, 0, stream>>>(nrm);
    coef_kernel<<<(Bc * FP + 255) / 256, 256, 0, stream>>>(tverts, faces, coef);
    raster_kernel<<<Bc * BLOCKS_PER_B, 256, 0, stream>>>(coef, tri);
    shade_kernel<<<(Bc * Hh * Ww + 255) / 256, 256, 0, stream>>>(
        albedos, lights, faces, uvcoords, cf, nrm, coef, tri, out);
}